// NMSLayer_8770323218833
// MI455X (gfx1250) — compile-verified
//
#include <hip/hip_runtime.h>

// ---------------- workspace layout (uint32 slots) ----------------
#define WS_HIST1 0        // 4096 bins (key[31:20])
#define WS_HIST2 4096     // 1024 bins (key[19:10])
#define WS_HIST3 5120     // 1024 bins (key[9:0])
#define WS_STATE 6144     // [0]=selected key prefix, [1]=remaining rank
#define WS_MED   6146     // final median (float bits)

#define TILE    64
#define IN_ROWS 70        // 64 + 2*3 halo
#define IN_COLS 72        // 18 x float4 (covers gx0-4 .. gx0+67)
#define NEG_INF (-__builtin_inff())

// -------- CDNA5 async vmem->LDS path probe --------
#if __has_builtin(__builtin_amdgcn_global_load_async_to_lds_b128)
#define HAVE_ASYNC 1
#else
#define HAVE_ASYNC 0
#if defined(__HIP_DEVICE_COMPILE__)
#warning "gfx1250 async_to_lds builtin unavailable - falling back to synchronous LDS staging"
#endif
#endif

// 16-byte int vector type matching the builtin's parameter type
typedef int v4i __attribute__((vector_size(4 * sizeof(int))));

__device__ __forceinline__ unsigned int fkey(float f) {
  unsigned int u = __float_as_uint(f);
  return (u & 0x80000000u) ? ~u : (u | 0x80000000u);
}

// ============ fused threshold + 7x7 maxpool + NMS (first in module) ============
__global__ __launch_bounds__(256)
void nms_pool(const float* __restrict__ x, float* __restrict__ out,
              const float* __restrict__ medPtr) {
  __shared__ float raw[IN_ROWS][IN_COLS];   // raw halo tile (col 0 == gx0-4)
  __shared__ float rmax[IN_ROWS][TILE];     // horizontal 7-max

  const int img = blockIdx.z;
  const int gy0 = blockIdx.y * TILE;
  const int gx0 = blockIdx.x * TILE;
  const float* __restrict__ base = x + (size_t)img * 1024u * 1024u;
  const float med = *medPtr;
  const int tid = threadIdx.x;

  // ---- Stage 70x72 tile via async vmem->LDS DMA (B128 per lane). ----
  // Addresses are clamped into the image so every load is legal & 16B-aligned;
  // a 4-float group is always entirely in- or out-of-image (gx0 % 64 == 0),
  // and out-of-image cells are replaced by -inf at consume time.
  for (int idx = tid; idx < IN_ROWS * (IN_COLS / 4); idx += 256) {
    int row = idx / (IN_COLS / 4);
    int g   = idx % (IN_COLS / 4);
    int gy = gy0 - 3 + row; gy = min(max(gy, 0), 1023);
    int cs = gx0 - 4 + 4 * g; cs = min(max(cs, 0), 1020);
    const float* gp = base + gy * 1024 + cs;
    float* lp = &raw[row][4 * g];
#if HAVE_ASYNC
    __builtin_amdgcn_global_load_async_to_lds_b128(
        (__attribute__((address_space(1))) v4i*)gp,
        (__attribute__((address_space(3))) v4i*)lp, 0, 0);
#else
    *(float4*)lp = *(const float4*)gp;
#endif
  }
#if HAVE_ASYNC
#if __has_builtin(__builtin_amdgcn_s_wait_asynccnt)
  __builtin_amdgcn_s_wait_asynccnt(0);
#else
  asm volatile("s_wait_asynccnt 0" ::: "memory");
#endif
#endif
  __syncthreads();

  // ---- Horizontal 7-max of thresholded values (with -inf outside image) ----
  for (int idx = tid; idx < IN_ROWS * TILE; idx += 256) {
    int row = idx >> 6;
    int xo  = idx & 63;
    int gy = gy0 - 3 + row;
    float m = NEG_INF;
    if ((unsigned)gy < 1024u) {
#pragma unroll
      for (int k = 0; k < 7; ++k) {
        int gx = gx0 + xo - 3 + k;
        float v = raw[row][xo + 1 + k];
        v = (v > med) ? v : 0.0f;                      // threshold below median
        m = fmaxf(m, ((unsigned)gx < 1024u) ? v : NEG_INF);
      }
    }
    rmax[row][xo] = m;
  }
  __syncthreads();

  // ---- Vertical 7-max + NMS equality mask, coalesced NT stores ----
  for (int idx = tid; idx < TILE * TILE; idx += 256) {
    int yo = idx >> 6;
    int xo = idx & 63;
    float m = NEG_INF;
#pragma unroll
    for (int r = 0; r < 7; ++r) m = fmaxf(m, rmax[yo + r][xo]);
    float c  = raw[yo + 3][xo + 4];                    // center (always in-image)
    float xt = (c > med) ? c : 0.0f;
    float r  = (xt == m) ? xt : 0.0f;
    __builtin_nontemporal_store(
        r, &out[(size_t)img * 1024u * 1024u + (size_t)(gy0 + yo) * 1024u + (gx0 + xo)]);
  }
}

// ============ radix-select median machinery ============
__global__ void init_ws(unsigned int* ws, unsigned int target) {
  int tid = blockIdx.x * blockDim.x + threadIdx.x;
  for (int i = tid; i < WS_STATE; i += blockDim.x * gridDim.x) ws[i] = 0u;
  if (tid == 0) { ws[WS_STATE] = 0u; ws[WS_STATE + 1] = target; ws[WS_MED] = 0u; }
}

__global__ __launch_bounds__(256)
void hist_pass(const float4* __restrict__ x4, unsigned int n4,
               unsigned int* __restrict__ hist,
               const unsigned int* __restrict__ state,
               unsigned int matchShift, unsigned int binShift,
               unsigned int binMask, int nbins) {
  __shared__ unsigned int lh[4096];
  for (int i = threadIdx.x; i < nbins; i += blockDim.x) lh[i] = 0u;
  __syncthreads();
  const unsigned int sel = state[0];
  const unsigned int stride = blockDim.x * gridDim.x;
  for (unsigned int i = blockIdx.x * blockDim.x + threadIdx.x; i < n4; i += stride) {
    __builtin_prefetch(&x4[i + stride], 0, 0);   // global_prefetch_b8
    float4 v = x4[i];
    unsigned int k;
    k = fkey(v.x); if (matchShift >= 32u || (k >> matchShift) == sel) atomicAdd(&lh[(k >> binShift) & binMask], 1u);
    k = fkey(v.y); if (matchShift >= 32u || (k >> matchShift) == sel) atomicAdd(&lh[(k >> binShift) & binMask], 1u);
    k = fkey(v.z); if (matchShift >= 32u || (k >> matchShift) == sel) atomicAdd(&lh[(k >> binShift) & binMask], 1u);
    k = fkey(v.w); if (matchShift >= 32u || (k >> matchShift) == sel) atomicAdd(&lh[(k >> binShift) & binMask], 1u);
  }
  __syncthreads();
  for (int i = threadIdx.x; i < nbins; i += blockDim.x) {
    unsigned int c = lh[i];
    if (c) atomicAdd(&hist[i], c);
  }
}

__global__ void select_bin(const unsigned int* __restrict__ hist, int nbins,
                           unsigned int* __restrict__ state, int shiftBits,
                           int isFinal, float* __restrict__ medOut) {
  if (threadIdx.x != 0 || blockIdx.x != 0) return;
  unsigned long long target = state[1];
  unsigned long long cum = 0;
  unsigned int b = 0;
  for (int i = 0; i < nbins; ++i) {
    unsigned long long h = hist[i];
    if (cum + h > target) { b = (unsigned int)i; break; }
    cum += h;
  }
  state[0] = (state[0] << shiftBits) | b;
  state[1] = (unsigned int)(target - cum);
  if (isFinal) {
    unsigned int k = state[0];
    unsigned int u = (k & 0x80000000u) ? (k & 0x7fffffffu) : ~k;
    *medOut = __uint_as_float(u);
  }
}

extern "C" void kernel_launch(void* const* d_in, const int* in_sizes, int n_in,
                              void* d_out, int out_size, void* d_ws, size_t ws_size,
                              hipStream_t stream) {
  const float* x = (const float*)d_in[0];
  float* out = (float*)d_out;
  const unsigned int n = (unsigned int)in_sizes[0];        // 16*1024*1024
  unsigned int* ws = (unsigned int*)d_ws;
  unsigned int* hist1 = ws + WS_HIST1;
  unsigned int* hist2 = ws + WS_HIST2;
  unsigned int* hist3 = ws + WS_HIST3;
  unsigned int* state = ws + WS_STATE;
  float* med = (float*)(ws + WS_MED);
  const unsigned int target = (n - 1u) / 2u;               // lower median rank

  init_ws<<<24, 256, 0, stream>>>(ws, target);

  const float4* x4 = (const float4*)x;
  const unsigned int n4 = n >> 2;
  // 3-pass radix select: 12 + 10 + 10 key bits
  hist_pass<<<2048, 256, 0, stream>>>(x4, n4, hist1, state, 32u, 20u, 0xFFFu, 4096);
  select_bin<<<1, 1, 0, stream>>>(hist1, 4096, state, 12, 0, med);
  hist_pass<<<2048, 256, 0, stream>>>(x4, n4, hist2, state, 20u, 10u, 0x3FFu, 1024);
  select_bin<<<1, 1, 0, stream>>>(hist2, 1024, state, 10, 0, med);
  hist_pass<<<2048, 256, 0, stream>>>(x4, n4, hist3, state, 10u, 0u, 0x3FFu, 1024);
  select_bin<<<1, 1, 0, stream>>>(hist3, 1024, state, 10, 1, med);

  dim3 grid(1024 / TILE, 1024 / TILE, 16);
  nms_pool<<<grid, 256, 0, stream>>>(x, out, med);
}